// SchurDecompositionLinear_24086176596621
// MI455X (gfx1250) — compile-verified
//
#include <hip/hip_runtime.h>

typedef float v2f __attribute__((ext_vector_type(2)));
typedef float v4f __attribute__((ext_vector_type(4)));
typedef float v8f __attribute__((ext_vector_type(8)));

#define NMAT 256

// ---------------------------------------------------------------------------
// Kernel 1: apply the Householder chain to M = [T ; I]  (512 x 256 rows).
// Rows are independent given u_i, so 16 blocks x 8 waves each own 4 rows in
// VGPRs for the whole 256-step serial chain. u_i broadcast through LDS.
// ---------------------------------------------------------------------------
__global__ __launch_bounds__(256) void hh_apply_kernel(
    const float* __restrict__ U, const float* __restrict__ theta,
    const float* __restrict__ gam, float* __restrict__ PT,
    float* __restrict__ PW)
{
  __shared__ float u_sh[NMAT];
  __shared__ float red[8];
  __shared__ float unorm_sh;

  const int t    = threadIdx.x;
  const int lane = t & 31;
  const int wave = t >> 5;

  float m[4][8];
  const int rbase = blockIdx.x * 32 + wave * 4;   // global row in [0,512)

  #pragma unroll
  for (int rr = 0; rr < 4; ++rr) {
    const int r = rbase + rr;
    float cg = 0.0f, sg = 0.0f;
    if (r < NMAT) {
      const int k = r >> 1;
      cg = gam[k] * cosf(theta[k]);
      sg = gam[k] * sinf(theta[k]);
    }
    #pragma unroll
    for (int j = 0; j < 8; ++j) {
      const int c = lane + 32 * j;
      float v = 0.0f;
      if (r < NMAT) {
        if ((r & 1) == 0) { if (c == r) v = cg; else if (c == r + 1) v = -sg; }
        else             { if (c == r - 1) v = sg; else if (c == r) v = cg; }
      } else {
        v = (c == (r - NMAT)) ? 1.0f : 0.0f;
      }
      m[rr][j] = v;
    }
  }

  for (int i = 0; i < NMAT; ++i) {
    __syncthreads();                       // protect u_sh from prior readers
    const float uv = (t >= i) ? U[i * NMAT + t] : 0.0f;
    u_sh[t] = uv;
    float p = uv * uv;
    #pragma unroll
    for (int off = 16; off > 0; off >>= 1) p += __shfl_xor(p, off, 32);
    if (lane == 0) red[wave] = p;
    __syncthreads();
    if (t == 0) {
      float s = 0.0f;
      #pragma unroll
      for (int w = 0; w < 8; ++w) s += red[w];
      unorm_sh = s;
    }
    __syncthreads();
    const float two_over = 2.0f / unorm_sh;
    #pragma unroll
    for (int rr = 0; rr < 4; ++rr) {
      float dot = 0.0f;
      #pragma unroll
      for (int j = 0; j < 8; ++j) dot += m[rr][j] * u_sh[lane + 32 * j];
      #pragma unroll
      for (int off = 16; off > 0; off >>= 1) dot += __shfl_xor(dot, off, 32);
      const float alpha = dot * two_over;
      #pragma unroll
      for (int j = 0; j < 8; ++j) m[rr][j] -= alpha * u_sh[lane + 32 * j];
    }
  }

  #pragma unroll
  for (int rr = 0; rr < 4; ++rr) {
    const int r = rbase + rr;
    float* dst = (r < NMAT) ? (PT + r * NMAT) : (PW + (r - NMAT) * NMAT);
    #pragma unroll
    for (int j = 0; j < 8; ++j) dst[lane + 32 * j] = m[rr][j];
  }
}

// ---------------------------------------------------------------------------
// Kernel 2: W = PT @ PW^T via V_WMMA_F32_16X16X4_F32, stored transposed
// (Wt[n][m] = W[m][n]) so the big GEMM can stage it contiguously.
// NT layout => both A and B fragments are contiguous b64 loads per lane.
// ---------------------------------------------------------------------------
__global__ __launch_bounds__(256) void wmat_kernel(
    const float* __restrict__ PT, const float* __restrict__ PW,
    float* __restrict__ Wt)
{
  const int lane = threadIdx.x & 31;
  const int wave = threadIdx.x >> 5;
  const int lo   = lane & 15;
  const int hi   = lane >> 4;
  const int tile = blockIdx.x * 8 + wave;        // 256 tiles total
  const int m0   = (tile >> 4) * 16;
  const int n0   = (tile & 15) * 16;

  // A layout (ISA): v0 = A[m, k + 2*hi], v1 = A[m, k + 2*hi + 1], m = m0+lo.
  const float* aptr = PT + (m0 + lo) * NMAT + 2 * hi;
  // B[k][n] = PW[n][k]  =>  contiguous along k for fixed n.
  const float* bptr = PW + (n0 + lo) * NMAT + 2 * hi;

  v8f acc = {};
  #pragma unroll 4
  for (int k = 0; k < NMAT; k += 4) {
    const v2f a = *(const v2f*)(aptr + k);
    const v2f b = *(const v2f*)(bptr + k);
    acc = __builtin_amdgcn_wmma_f32_16x16x4_f32(false, a, false, b,
                                                (short)0, acc, false, false);
  }
  // C layout: VGPR v -> row m0 + v + 8*hi, col n0 + lo.  Store transposed.
  #pragma unroll
  for (int v = 0; v < 8; ++v)
    Wt[(n0 + lo) * NMAT + (m0 + v + 8 * hi)] = acc[v];
}

// ---------------------------------------------------------------------------
// Kernel 3: out = x @ W  (65536x256 @ 256x256), WMMA f32.
// Each wave owns 16 rows; all 64 A-fragments preloaded into VGPRs so x is
// read exactly once. Wt staged into LDS in 32-column chunks with padded
// stride 260 (conflict-free fills, bank-clean b64 B-fragment reads).
// ---------------------------------------------------------------------------
#define NCHUNK 32
#define LPAD   260

__global__ __launch_bounds__(256) void xw_gemm_kernel(
    const float* __restrict__ X, const float* __restrict__ Wt,
    float* __restrict__ out)
{
  __shared__ __align__(16) float wl[NCHUNK * LPAD];   // ~33 KB

  const int t    = threadIdx.x;
  const int lane = t & 31;
  const int wave = t >> 5;
  const int lo   = lane & 15;
  const int hi   = lane >> 4;
  const int m0   = blockIdx.x * 128 + wave * 16;

  // Preload all A fragments for this wave's 16 rows (K = 256 -> 64 steps).
  v2f afrag[64];
  const float* aptr = X + (size_t)(m0 + lo) * NMAT + 2 * hi;
  #pragma unroll
  for (int kk = 0; kk < 64; ++kk)
    afrag[kk] = *(const v2f*)(aptr + kk * 4);

  for (int p = 0; p < NMAT / NCHUNK; ++p) {
    const int n0 = p * NCHUNK;
    __syncthreads();
    // Fill: wl[nl][k] = W[k][n0+nl] = Wt[(n0+nl)*N + k]; coalesced float4.
    #pragma unroll
    for (int it = 0; it < 8; ++it) {
      const int e  = it * 256 + t;       // 2048 float4 elements
      const int nl = e >> 6;
      const int k4 = (e & 63) * 4;
      *(v4f*)(&wl[nl * LPAD + k4]) = *(const v4f*)(Wt + (n0 + nl) * NMAT + k4);
    }
    __syncthreads();

    v8f acc0 = {}, acc1 = {};
    #pragma unroll
    for (int kk = 0; kk < 64; ++kk) {
      const v2f a  = afrag[kk];
      const v2f b0 = *(const v2f*)(&wl[(lo)      * LPAD + kk * 4 + 2 * hi]);
      const v2f b1 = *(const v2f*)(&wl[(16 + lo) * LPAD + kk * 4 + 2 * hi]);
      acc0 = __builtin_amdgcn_wmma_f32_16x16x4_f32(false, a, false, b0,
                                                   (short)0, acc0, false, false);
      acc1 = __builtin_amdgcn_wmma_f32_16x16x4_f32(false, a, false, b1,
                                                   (short)0, acc1, false, false);
    }
    #pragma unroll
    for (int v = 0; v < 8; ++v) {
      const int row = m0 + v + 8 * hi;
      out[(size_t)row * NMAT + n0 + lo]      = acc0[v];
      out[(size_t)row * NMAT + n0 + 16 + lo] = acc1[v];
    }
  }
}

// ---------------------------------------------------------------------------
extern "C" void kernel_launch(void* const* d_in, const int* in_sizes, int n_in,
                              void* d_out, int out_size, void* d_ws,
                              size_t ws_size, hipStream_t stream) {
  (void)in_sizes; (void)n_in; (void)out_size; (void)ws_size;
  const float* x     = (const float*)d_in[0];  // [65536, 256]
  const float* U     = (const float*)d_in[1];  // [256, 256]
  const float* theta = (const float*)d_in[2];  // [128]
  const float* gam   = (const float*)d_in[3];  // [128]
  float* out = (float*)d_out;

  float* ws = (float*)d_ws;
  float* PT = ws;                 // 256*256
  float* PW = ws + 65536;         // 256*256
  float* Wt = ws + 131072;        // 256*256 (W transposed)

  hh_apply_kernel<<<16, 256, 0, stream>>>(U, theta, gam, PT, PW);
  wmat_kernel<<<32, 256, 0, stream>>>(PT, PW, Wt);
  xw_gemm_kernel<<<512, 256, 0, stream>>>(x, Wt, out);
}